// GAT_29102698398114
// MI455X (gfx1250) — compile-verified
//
#include <hip/hip_runtime.h>
#include <hip/hip_bf16.h>

#define NEG_SLOPE 0.2f
#define GN 64          // inner feature width (H*C) for both layers
#define BM 128         // GEMM rows per block (8 waves x 16)
#define KC 32          // GEMM K chunk
#define SBSTRIDE 160   // pair-row stride: 160 mod 64 == 32 -> half-waves on disjoint banks

typedef float v2f __attribute__((ext_vector_type(2)));
typedef float v8f __attribute__((ext_vector_type(8)));

// ---------------------------------------------------------------------------
// GEMM: C[M,64] = A[M,K] * B[K,64], fp32 via V_WMMA_F32_16X16X4_F32.
// 256 threads = 8 waves; each wave owns a 16-row M tile and all 4 N tiles.
// A fragment (16x4): lanes 0-15 -> M=lane,    v0/v1 = K {kk+0, kk+1};
//                    lanes16-31 -> M=lane-16, v0/v1 = K {kk+2, kk+3}.
// B is staged K-pair-interleaved: sBp[p][2n..2n+1] = {B[2p][n], B[2p+1][n]}
// so each B fragment is one contiguous ds_load_b64.
// C/D layout: VGPR r -> M=r (lanes 0-15) / M=r+8 (lanes 16-31), N = lane&15.
// ---------------------------------------------------------------------------
__global__ __launch_bounds__(256) void gemm_n64_wmma(
    const float* __restrict__ A, const float* __restrict__ B,
    float* __restrict__ C, int M, int K) {
  __shared__ float sA[BM][34];              // stride 34: conflict-free b64 reads
  __shared__ float sBp[KC / 2][SBSTRIDE];   // K-pair interleaved B tile

  const int t    = threadIdx.x;
  const int lane = t & 31;
  const int wave = t >> 5;
  const int rowBase = blockIdx.x * BM;

  v8f acc0 = {}, acc1 = {}, acc2 = {}, acc3 = {};
  const int kb = (lane >> 4) << 1;   // 0 (lo half) / 2 (hi half)
  const int hn = lane & 15;

  // Per-thread A-load coordinates (reused for the global_prefetch of chunk k0+KC).
  const int aR0 = t >> 3;            // row of first float4 this thread loads
  const int aC0 = (t & 7) << 2;      // col of first float4 this thread loads

  for (int k0 = 0; k0 < K; k0 += KC) {
    // Cooperative A tile load: BM x KC, coalesced float4.
#pragma unroll
    for (int i = 0; i < 4; ++i) {
      int idx = t + i * 256;              // 0..1023
      int r   = idx >> 3;
      int c4  = (idx & 7) << 2;
      int gr  = rowBase + r;
      float4 v = make_float4(0.f, 0.f, 0.f, 0.f);
      if (gr < M) v = *(const float4*)(A + (long long)gr * K + k0 + c4);
      sA[r][c4 + 0] = v.x; sA[r][c4 + 1] = v.y;
      sA[r][c4 + 2] = v.z; sA[r][c4 + 3] = v.w;
    }
    // Cooperative B tile load: KC x 64, stored K-pair interleaved.
#pragma unroll
    for (int i = 0; i < 2; ++i) {
      int idx = t + i * 256;              // 0..511
      int r    = idx >> 4;                // K row within chunk
      int c4   = (idx & 15) << 2;         // N col
      int p    = r >> 1;
      int half = r & 1;
      float4 v = *(const float4*)(B + (long long)(k0 + r) * GN + c4);
      float* dp = &sBp[p][0];
      dp[(c4 + 0) * 2 + half] = v.x;
      dp[(c4 + 1) * 2 + half] = v.y;
      dp[(c4 + 2) * 2 + half] = v.z;
      dp[(c4 + 3) * 2 + half] = v.w;
    }
    __syncthreads();

    // Prefetch next A chunk (dominant 205MB stream) while this chunk computes.
    if (k0 + KC < K) {
      int gr = rowBase + aR0;
      if (gr < M)
        __builtin_prefetch(A + (long long)gr * K + k0 + KC + aC0, 0, 0);
    }

    const float* aRow   = &sA[wave * 16 + hn][0];
    const int    colOff = hn << 1;
#pragma unroll
    for (int kk = 0; kk < KC; kk += 4) {
      int p = (kk + kb) >> 1;
      v2f a = *(const v2f*)(aRow + kk + kb);          // ds_load_b64
      const float* bp = &sBp[p][colOff];
      v2f b0 = *(const v2f*)(bp);                     // ds_load_b64 each
      v2f b1 = *(const v2f*)(bp + 32);
      v2f b2 = *(const v2f*)(bp + 64);
      v2f b3 = *(const v2f*)(bp + 96);
      acc0 = __builtin_amdgcn_wmma_f32_16x16x4_f32(false, a, false, b0, (short)0, acc0, false, false);
      acc1 = __builtin_amdgcn_wmma_f32_16x16x4_f32(false, a, false, b1, (short)0, acc1, false, false);
      acc2 = __builtin_amdgcn_wmma_f32_16x16x4_f32(false, a, false, b2, (short)0, acc2, false, false);
      acc3 = __builtin_amdgcn_wmma_f32_16x16x4_f32(false, a, false, b3, (short)0, acc3, false, false);
    }
    __syncthreads();
  }

  // Store: VGPR r holds row r (lanes 0-15) / r+8 (lanes 16-31).
  int mBase = rowBase + wave * 16 + ((lane >> 4) << 3);
#pragma unroll
  for (int r = 0; r < 8; ++r) {
    int gm = mBase + r;
    if (gm < M) {
      float* crow = C + (long long)gm * GN + hn;
      crow[0]  = acc0[r];
      crow[16] = acc1[r];
      crow[32] = acc2[r];
      crow[48] = acc3[r];
    }
  }
}

// ---------------------------------------------------------------------------
// Per-(node, head) attention logits: al = <xh[n,h,:], a[h,:]>
// ---------------------------------------------------------------------------
__global__ void attn_prep_k(const float* __restrict__ xh,
                            const float* __restrict__ asrc,
                            const float* __restrict__ adst,
                            float* __restrict__ als, float* __restrict__ ald,
                            int N, int H, int C) {
  int t = blockIdx.x * blockDim.x + threadIdx.x;
  int node = t / H;
  int h    = t - node * H;
  if (node >= N) return;
  const float* row = xh + (long long)node * GN + h * C;
  const float* as  = asrc + h * C;
  const float* ad  = adst + h * C;
  float ss = 0.f, sd = 0.f;
  for (int c = 0; c < C; ++c) {
    float x = row[c];
    ss += x * as[c];
    sd += x * ad[c];
  }
  als[t] = ss;
  ald[t] = sd;
}

// Zero accumulators / init segment stats (emax=-inf, den=0).
__global__ void seg_init_k(float* __restrict__ acc, float* __restrict__ emax,
                           float* __restrict__ den, int n64, int nH) {
  int t = blockIdx.x * blockDim.x + threadIdx.x;
  if (t < n64) acc[t] = 0.f;
  if (t < nH) {
    emax[t] = __int_as_float(0xFF800000);  // -inf
    den[t]  = 0.f;
  }
}

__device__ __forceinline__ void edge_nodes(const int* __restrict__ ei, int e,
                                           int E, int& s, int& d) {
  if (e < E) { s = ei[e]; d = ei[E + e]; }
  else       { s = e - E; d = s; }        // appended self loops
}

__device__ __forceinline__ float leaky(float v) {
  return v >= 0.f ? v : NEG_SLOPE * v;
}

// Float atomic max via signed/unsigned int trick (init must be -inf).
__device__ __forceinline__ void atomicMaxF(float* addr, float v) {
  if (v >= 0.f) atomicMax((int*)addr, __float_as_int(v));
  else          atomicMin((unsigned int*)addr, __float_as_uint(v));
}

// Pass A: segment max of leaky-relu(al_src[src]+al_dst[dst]) over dst.
__global__ void edge_max_k(const int* __restrict__ ei,
                           const float* __restrict__ als,
                           const float* __restrict__ ald,
                           float* __restrict__ emax, int E, int Etot, int H) {
  int t = blockIdx.x * blockDim.x + threadIdx.x;
  int e = t / H, h = t - e * H;
  if (e >= Etot) return;
  int s, d; edge_nodes(ei, e, E, s, d);
  float v = leaky(als[s * H + h] + ald[d * H + h]);
  atomicMaxF(&emax[d * H + h], v);
}

// Pass B: den[dst] += exp(e - emax[dst])
__global__ void edge_den_k(const int* __restrict__ ei,
                           const float* __restrict__ als,
                           const float* __restrict__ ald,
                           const float* __restrict__ emax,
                           float* __restrict__ den, int E, int Etot, int H) {
  int t = blockIdx.x * blockDim.x + threadIdx.x;
  int e = t / H, h = t - e * H;
  if (e >= Etot) return;
  int s, d; edge_nodes(ei, e, E, s, d);
  int ai = d * H + h;
  float v = leaky(als[s * H + h] + ald[ai]);
  atomicAdd(&den[ai], __expf(v - emax[ai]));
}

// Pass C: out[dst, j] += alpha * xh[src, j]; 64 threads per edge (coalesced).
__global__ void edge_agg_k(const int* __restrict__ ei,
                           const float* __restrict__ als,
                           const float* __restrict__ ald,
                           const float* __restrict__ emax,
                           const float* __restrict__ den,
                           const float* __restrict__ xh,
                           float* __restrict__ acc,
                           int E, int Etot, int H, int log2C) {
  int t = blockIdx.x * blockDim.x + threadIdx.x;
  int e = t >> 6, j = t & 63;
  if (e >= Etot) return;
  int head = j >> log2C;
  int s, d; edge_nodes(ei, e, E, s, d);
  int ai = d * H + head;
  float v     = leaky(als[s * H + head] + ald[ai]);
  float alpha = __expf(v - emax[ai]) / (den[ai] + 1e-16f);
  atomicAdd(&acc[(long long)d * GN + j], xh[(long long)s * GN + j] * alpha);
}

// h = relu(acc + b)
__global__ void bias_relu_k(const float* __restrict__ acc,
                            const float* __restrict__ b,
                            float* __restrict__ out, int n64) {
  int t = blockIdx.x * blockDim.x + threadIdx.x;
  if (t >= n64) return;
  float v = acc[t] + b[t & 63];
  out[t] = v > 0.f ? v : 0.f;
}

// In-place: io[n,:] = log_softmax(io[n,:] + b); one wave32 per node.
__global__ void bias_logsoftmax_k(float* __restrict__ io,
                                  const float* __restrict__ b, int N) {
  int gid  = blockIdx.x * blockDim.x + threadIdx.x;
  int node = gid >> 5;
  int lane = gid & 31;
  if (node >= N) return;
  float v0 = io[(long long)node * GN + lane]      + b[lane];
  float v1 = io[(long long)node * GN + lane + 32] + b[lane + 32];
  float m = fmaxf(v0, v1);
#pragma unroll
  for (int o = 16; o > 0; o >>= 1) m = fmaxf(m, __shfl_xor(m, o, 32));
  float s = __expf(v0 - m) + __expf(v1 - m);
#pragma unroll
  for (int o = 16; o > 0; o >>= 1) s += __shfl_xor(s, o, 32);
  float lse = m + __logf(s);
  io[(long long)node * GN + lane]      = v0 - lse;
  io[(long long)node * GN + lane + 32] = v1 - lse;
}

// ---------------------------------------------------------------------------
extern "C" void kernel_launch(void* const* d_in, const int* in_sizes, int n_in,
                              void* d_out, int out_size, void* d_ws, size_t ws_size,
                              hipStream_t stream) {
  const float* x   = (const float*)d_in[0];
  const int*   ei  = (const int*)  d_in[1];
  const float* W1  = (const float*)d_in[2];
  const float* a1s = (const float*)d_in[3];
  const float* a1d = (const float*)d_in[4];
  const float* b1  = (const float*)d_in[5];
  const float* W2  = (const float*)d_in[6];
  const float* a2s = (const float*)d_in[7];
  const float* a2d = (const float*)d_in[8];
  const float* b2  = (const float*)d_in[9];
  float* out = (float*)d_out;

  const int K1 = 512;
  const int N    = in_sizes[0] / K1;
  const int E    = in_sizes[1] / 2;
  const int Etot = E + N;               // edges + self loops

  float* ws  = (float*)d_ws;
  float* xh  = ws; ws += (size_t)N * 64;   // layer features (reused layer 2)
  float* h   = ws; ws += (size_t)N * 64;   // relu output of layer 1
  float* acc = ws; ws += (size_t)N * 64;   // layer-1 aggregation buffer
  float* als = ws; ws += (size_t)N * 8;
  float* ald = ws; ws += (size_t)N * 8;
  float* emx = ws; ws += (size_t)N * 8;
  float* den = ws; ws += (size_t)N * 8;

  dim3 blk(256);
  const unsigned gM   = (unsigned)((N + BM - 1) / BM);
  const unsigned gN64 = (unsigned)((N * 64 + 255) / 256);
  const unsigned gAgg = (unsigned)(((long long)Etot * 64 + 255) / 256);

  // ---------------- layer 1 (H=8, C=8) ----------------
  gemm_n64_wmma<<<gM, blk, 0, stream>>>(x, W1, xh, N, K1);
  attn_prep_k  <<<(unsigned)((N * 8 + 255) / 256), blk, 0, stream>>>(xh, a1s, a1d, als, ald, N, 8, 8);
  seg_init_k   <<<gN64, blk, 0, stream>>>(acc, emx, den, N * 64, N * 8);
  edge_max_k   <<<(unsigned)(((long long)Etot * 8 + 255) / 256), blk, 0, stream>>>(ei, als, ald, emx, E, Etot, 8);
  edge_den_k   <<<(unsigned)(((long long)Etot * 8 + 255) / 256), blk, 0, stream>>>(ei, als, ald, emx, den, E, Etot, 8);
  edge_agg_k   <<<gAgg, blk, 0, stream>>>(ei, als, ald, emx, den, xh, acc, E, Etot, 8, 3);
  bias_relu_k  <<<gN64, blk, 0, stream>>>(acc, b1, h, N * 64);

  // ---------------- layer 2 (H=1, C=64) ----------------
  gemm_n64_wmma<<<gM, blk, 0, stream>>>(h, W2, xh, N, 64);
  attn_prep_k  <<<(unsigned)((N + 255) / 256), blk, 0, stream>>>(xh, a2s, a2d, als, ald, N, 1, 64);
  seg_init_k   <<<gN64, blk, 0, stream>>>(out, emx, den, N * 64, N);
  edge_max_k   <<<(unsigned)((Etot + 255) / 256), blk, 0, stream>>>(ei, als, ald, emx, E, Etot, 1);
  edge_den_k   <<<(unsigned)((Etot + 255) / 256), blk, 0, stream>>>(ei, als, ald, emx, den, E, Etot, 1);
  edge_agg_k   <<<gAgg, blk, 0, stream>>>(ei, als, ald, emx, den, xh, out, E, Etot, 1, 6);
  bias_logsoftmax_k<<<(unsigned)(((long long)N * 32 + 255) / 256), blk, 0, stream>>>(out, b2, N);
}